// GraphConvolution_9758165697126
// MI455X (gfx1250) — compile-verified
//
#include <hip/hip_runtime.h>

typedef __attribute__((ext_vector_type(2))) float v2f;
typedef __attribute__((ext_vector_type(8))) float v8f;

#define N_NODES   50000
#define K_DIM     128
#define N_OUT     32
#define M_TILES   (N_NODES / 16)      // 3125, exact
#define WAVES_PB  8                   // 256 threads = 8 wave32
#define WT_PAD    132                 // transposed-W row stride (dwords); 132%64==4
                                      // -> wave32 b64 B-loads hit 64 distinct banks

// ---------------------------------------------------------------------------
// Kernel 1: h = x @ W  via V_WMMA_F32_16X16X4_F32 (fp32-exact, matches ref)
// One wave -> 16x32 tile of h; two v8f accumulators (N columns 0-15, 16-31).
// W is staged TRANSPOSED in LDS so each B fragment is one aligned ds_load_b64.
// ---------------------------------------------------------------------------
__global__ __launch_bounds__(256) void gemm_xw_wmma(const float* __restrict__ x,
                                                    const float* __restrict__ w,
                                                    float* __restrict__ h) {
    __shared__ float WlT[N_OUT * WT_PAD];    // ~16.9 KB of the 320 KB WGP LDS
    for (int i = threadIdx.x; i < K_DIM * N_OUT; i += 256) {
        int k = i >> 5;                      // 0..127
        int n = i & 31;                      // 0..31
        WlT[n * WT_PAD + k] = w[i];          // w is [K][N] row-major
    }
    __syncthreads();

    const int wave = threadIdx.x >> 5;
    const int lane = threadIdx.x & 31;
    const int tile = blockIdx.x * WAVES_PB + wave;
    if (tile >= M_TILES) return;             // wave-uniform: EXEC stays all-1s

    const int m0   = tile << 4;              // first of 16 rows
    const int half = lane >> 4;              // 0 = lanes 0-15, 1 = lanes 16-31
    const int l    = lane & 15;

    const float* __restrict__ arow = x + (size_t)(m0 + l) * K_DIM;
    const float* __restrict__ b0p  = &WlT[l * WT_PAD];          // column n = l
    const float* __restrict__ b1p  = &WlT[(16 + l) * WT_PAD];   // column n = 16+l

    v8f acc0 = {};                           // N = 0..15
    v8f acc1 = {};                           // N = 16..31

    #pragma unroll 4
    for (int k = 0; k < K_DIM; k += 4) {
        const int ka = k + half * 2;         // K-pair this lane-half owns
        // A frag (16x4 f32): VGPR0 = K{0|2}, VGPR1 = K{1|3}  -> global_load_b64
        v2f a = *(const v2f*)(arow + ka);
        // B frags (4x16 f32): rows ka, ka+1 at column n are contiguous in WlT
        // -> single aligned ds_load_b64 each, no register shuffling
        v2f b0 = *(const v2f*)(b0p + ka);
        v2f b1 = *(const v2f*)(b1p + ka);
        // 8-arg form: (neg_a, A, neg_b, B, c_mod, C, reuse_a, reuse_b)
        acc0 = __builtin_amdgcn_wmma_f32_16x16x4_f32(
                   false, a, false, b0, (short)0, acc0, false, false);
        acc1 = __builtin_amdgcn_wmma_f32_16x16x4_f32(
                   false, a, false, b1, (short)0, acc1, false, false);
    }

    // C/D layout: VGPR r holds rows {r | r+8} split at lane 16, N = l
    const int mb = m0 + half * 8;
    #pragma unroll
    for (int r = 0; r < 8; ++r) {
        h[(size_t)(mb + r) * N_OUT + l]      = acc0[r];
        h[(size_t)(mb + r) * N_OUT + 16 + l] = acc1[r];
    }
}

// ---------------------------------------------------------------------------
// Kernel 2: zero the output (harness poisons d_out; scatter accumulates)
// ---------------------------------------------------------------------------
__global__ void zero_out_kernel(float* __restrict__ out, int n) {
    int i = blockIdx.x * blockDim.x + threadIdx.x;
    if (i < n) out[i] = 0.0f;
}

// ---------------------------------------------------------------------------
// Kernel 3: COO scatter-add. One wave per edge, one lane per feature.
// Gather h[src] (128B coalesced), scale, GLOBAL_ATOMIC_ADD_F32 into out[dst].
// h and out are 6.4 MB each -> fully resident in the 192 MB L2, so the random
// gather/scatter is L2-bound, not HBM-bound.
// ---------------------------------------------------------------------------
__global__ __launch_bounds__(256) void spmm_scatter(const float* __restrict__ h,
                                                    const int*   __restrict__ esrc,
                                                    const int*   __restrict__ edst,
                                                    const float* __restrict__ ew,
                                                    float* __restrict__ out,
                                                    int nE) {
    long long t = (long long)blockIdx.x * blockDim.x + threadIdx.x;
    int e = (int)(t >> 5);
    if (e >= nE) return;
    int j = (int)(t & 31);

    int   s  = esrc[e];          // wave-uniform -> scalar load
    int   d  = edst[e];
    float we = ew[e];

    float v = we * h[(size_t)s * N_OUT + j];
    atomicAdd(out + (size_t)d * N_OUT + j, v);
}

// ---------------------------------------------------------------------------
extern "C" void kernel_launch(void* const* d_in, const int* in_sizes, int n_in,
                              void* d_out, int out_size, void* d_ws, size_t ws_size,
                              hipStream_t stream) {
    const float* x    = (const float*)d_in[0];   // [50000,128] f32
    const float* w    = (const float*)d_in[1];   // [128,32]    f32
    const int*   esrc = (const int*)  d_in[2];   // [E]         i32
    const int*   edst = (const int*)  d_in[3];   // [E]         i32
    const float* ew   = (const float*)d_in[4];   // [E]         f32
    float*       out  = (float*)d_out;           // [50000,32]  f32
    float*       h    = (float*)d_ws;            // scratch: 6.4 MB

    const int nE = in_sizes[2];

    // 1) h = x @ W (WMMA f32)
    gemm_xw_wmma<<<(M_TILES + WAVES_PB - 1) / WAVES_PB, 256, 0, stream>>>(x, w, h);

    // 2) out = 0
    const int nOut = N_NODES * N_OUT;
    zero_out_kernel<<<(nOut + 255) / 256, 256, 0, stream>>>(out, nOut);

    // 3) scatter-add messages
    long long totalT = (long long)nE * N_OUT;
    int blocks = (int)((totalT + 255) / 256);
    spmm_scatter<<<blocks, 256, 0, stream>>>(h, esrc, edst, ew, out, nE);
}